// LatticeRNN_61649960567392
// MI455X (gfx1250) — compile-verified
//
#include <hip/hip_runtime.h>
#include <cmath>

#define N_NODES 32768
#define P_PAR   4
#define D_DIM   256
#define H_DIM   256

typedef __attribute__((ext_vector_type(2))) float v2f;
typedef __attribute__((ext_vector_type(4))) float v4f;
typedef __attribute__((ext_vector_type(8))) float v8f;

// ---------------------------------------------------------------------------
// Phase 1: xW[N,H] = emb[tags] @ W_ih^T + b_ih  via v_wmma_f32_16x16x4_f32.
// One wave computes one 16x16 output tile; K=256 -> 64 WMMA steps of K=4.
// A frag (16x4 f32): lane L holds A[M=L&15][K = k + 2*(L>>4) + {0,1}]  (2 VGPRs)
// B frag (4x16 f32): lane L holds B[K = k + 2*(L>>4) + {0,1}][N=L&15]  (2 VGPRs)
// B = W_ih^T, so B[k][n] = W_ih[n][k] -> contiguous float2 from W_ih row n.
// ---------------------------------------------------------------------------
__global__ __launch_bounds__(256) void input_proj_wmma(
    const int* __restrict__ tags, const float* __restrict__ emb,
    const float* __restrict__ W_ih, const float* __restrict__ b_ih,
    float* __restrict__ xW)
{
  const int lane  = threadIdx.x & 31;
  const int wave  = threadIdx.x >> 5;
  const int tile  = blockIdx.x * 8 + wave;   // 4096 blocks * 8 waves = 32768 tiles
  const int tileM = tile >> 4;               // 2048 row tiles
  const int tileN = tile & 15;               // 16 col tiles
  const int r     = lane & 15;
  const int koff  = (lane >> 4) << 1;        // 0 or 2

  const int gm = tileM * 16;
  const int gn = tileN * 16;

  const float* arow = emb  + (size_t)tags[gm + r] * D_DIM;  // embedding gather
  const float* brow = W_ih + (size_t)(gn + r)     * D_DIM;  // W_ih row = B column

  v8f c = {};
  #pragma unroll 8
  for (int k = 0; k < D_DIM; k += 4) {
    v2f a = *(const v2f*)(arow + k + koff);
    v2f b = *(const v2f*)(brow + k + koff);
    c = __builtin_amdgcn_wmma_f32_16x16x4_f32(false, a, false, b, (short)0, c,
                                              false, false);
  }

  // D layout: VGPR v -> (M = v + 8*(lane>=16), N = lane&15)
  const int n     = lane & 15;
  const int mbase = (lane >> 4) << 3;
  const float bias = b_ih[gn + n];
  #pragma unroll
  for (int v = 0; v < 8; ++v)
    xW[(size_t)(gm + mbase + v) * H_DIM + gn + n] = c[v] + bias;
}

// ---------------------------------------------------------------------------
// Phase 2: sequential lattice scan. Single persistent workgroup (1024 thr,
// 32 waves on one WGP). W_hh register-resident: thread (j = t&255, q = t>>8)
// holds W_hh[j][64q..64q+63] in 16 v4f regs. 64-entry hidden ring in LDS.
//
// Key property: every ring access is ring[slot*256 + j] by thread t==j
// (q==0) -> ring column j is thread-private through LDS, so NO barrier is
// needed between the ring write (end of step i) and the parent gather
// (start of step i+1). Only 2 barriers/step: z-publish and pred-publish.
// Global inputs for step i+1 (parents row, n_parents, xW column) are
// software-pipelined into the GEMV window of step i.
// ---------------------------------------------------------------------------
__global__ __launch_bounds__(1024) void lattice_scan(
    const int* __restrict__ parents, const int* __restrict__ n_parents,
    const float* __restrict__ xW, const float* __restrict__ W_hh,
    const float* __restrict__ b_hh, float* __restrict__ hs)
{
  extern __shared__ float smem[];
  float* ring = smem;              // [64][256] = 64 KB
  float* z    = ring + 64 * 256;   // [256]
  float* pred = z + 256;           // [1024]

  const int t = threadIdx.x;
  const int j = t & 255;           // output index
  const int q = t >> 8;            // k-quarter 0..3

  // register-resident weight slice (64 floats as 16 x v4f)
  v4f wv[16];
  {
    const v4f* wsrc = (const v4f*)(W_hh + (size_t)j * H_DIM + (q << 6));
    #pragma unroll
    for (int cidx = 0; cidx < 16; ++cidx) wv[cidx] = wsrc[cidx];
  }
  const float bj = b_hh[j];
  const int4* parents4 = (const int4*)parents;

  // pipeline registers for step 0 (q==0 threads only use these)
  int  np  = n_parents[0];
  int4 pv  = parents4[0];
  float xwv = xW[j];

  for (int i = 0; i < N_NODES; ++i) {
    // --- masked parent mean: 256 threads, one z element each (own column) --
    if (q == 0) {
      float acc = 0.f;
      if (np > 0) acc += ring[(pv.x & 63) * 256 + j];
      if (np > 1) acc += ring[(pv.y & 63) * 256 + j];
      if (np > 2) acc += ring[(pv.z & 63) * 256 + j];
      if (np > 3) acc += ring[(pv.w & 63) * 256 + j];
      z[j] = acc * (np > 0 ? 1.f / (float)np : 1.f);
    }
    __syncthreads();                               // z published

    // --- prefetch step i+1 inputs into the GEMV latency window -------------
    int np_n = 0; int4 pv_n; float xw_n = 0.f;
    if (q == 0 && i + 1 < N_NODES) {
      np_n = n_parents[i + 1];
      pv_n = parents4[i + 1];
      xw_n = xW[(size_t)(i + 1) * H_DIM + j];
    }
    if (q == 3 && i + 8 < N_NODES)                 // global_prefetch_b8 stream
      __builtin_prefetch(xW + (size_t)(i + 8) * H_DIM + j, 0, 2);

    // --- GEMV partial from register-resident weights -----------------------
    {
      const v4f* zq = (const v4f*)(z + (q << 6));  // lanes in a wave share q:
      float acc = 0.f;                             // ds_load_b128 broadcasts
      #pragma unroll
      for (int cidx = 0; cidx < 16; ++cidx) {
        v4f zv = zq[cidx];
        acc += wv[cidx].x * zv.x + wv[cidx].y * zv.y
             + wv[cidx].z * zv.z + wv[cidx].w * zv.w;
      }
      pred[t] = acc;
    }
    __syncthreads();                               // pred published

    // --- reduce 4 quarters, activate, publish (own ring column) ------------
    if (q == 0) {
      const float pre = xwv + pred[j] + pred[256 + j] + pred[512 + j]
                      + pred[768 + j] + bj;
      const float hv = tanhf(pre);
      ring[(i & 63) * 256 + j] = hv;               // thread-private column
      hs[(size_t)i * H_DIM + j] = hv;              // fire-and-forget
      np = np_n; pv = pv_n; xwv = xw_n;            // advance pipeline
    }
    // no third barrier: ring column j is only touched by thread j (q==0);
    // z/pred WAR hazards are covered by the two barriers of the next step.
  }
}

// ---------------------------------------------------------------------------
// Phase 3: out = sigmoid(hs @ W_out^T + b_out). One wave per row.
// ---------------------------------------------------------------------------
__global__ __launch_bounds__(256) void output_proj(
    const float* __restrict__ hs, const float* __restrict__ W_out,
    const float* __restrict__ b_out, float* __restrict__ out)
{
  const int lane = threadIdx.x & 31;
  const int wave = threadIdx.x >> 5;
  const int row  = blockIdx.x * 8 + wave;

  const float* hrow = hs + (size_t)row * H_DIM;
  float acc = 0.f;
  #pragma unroll
  for (int s = 0; s < 8; ++s)
    acc += hrow[lane + 32 * s] * W_out[lane + 32 * s];

  #pragma unroll
  for (int off = 16; off > 0; off >>= 1)
    acc += __shfl_xor(acc, off, 32);

  if (lane == 0)
    out[row] = 1.f / (1.f + expf(-(acc + b_out[0])));
}

// ---------------------------------------------------------------------------
extern "C" void kernel_launch(void* const* d_in, const int* in_sizes, int n_in,
                              void* d_out, int out_size, void* d_ws, size_t ws_size,
                              hipStream_t stream) {
  const int*   tags      = (const int*)  d_in[0];
  const int*   parents   = (const int*)  d_in[1];
  const int*   n_parents = (const int*)  d_in[2];
  const float* emb       = (const float*)d_in[3];
  const float* W_ih      = (const float*)d_in[4];
  const float* b_ih      = (const float*)d_in[5];
  const float* W_hh      = (const float*)d_in[6];
  const float* b_hh      = (const float*)d_in[7];
  const float* W_out     = (const float*)d_in[8];
  const float* b_out     = (const float*)d_in[9];
  float*       out       = (float*)d_out;

  float* xW = (float*)d_ws;                          // 32 MB
  float* hs = xW + (size_t)N_NODES * H_DIM;          // 32 MB

  // Phase 1: WMMA input projection (32768 tiles, 8 waves/block)
  input_proj_wmma<<<4096, 256, 0, stream>>>(tags, emb, W_ih, b_ih, xW);

  // Phase 2: persistent single-WGP sequential scan, 69 KB dynamic LDS
  const size_t smem_bytes = (64 * 256 + 256 + 1024) * sizeof(float);
  hipFuncSetAttribute(reinterpret_cast<const void*>(lattice_scan),
                      hipFuncAttributeMaxDynamicSharedMemorySize,
                      (int)smem_bytes);
  lattice_scan<<<1, 1024, smem_bytes, stream>>>(parents, n_parents, xW, W_hh,
                                                b_hh, hs);

  // Phase 3: output GEMV + sigmoid
  output_proj<<<4096, 256, 0, stream>>>(hs, W_out, b_out, out);
}